// GCN_86388972191750
// MI455X (gfx1250) — compile-verified
//
#include <hip/hip_runtime.h>
#include <hip/hip_bf16.h>

// ---------------------------------------------------------------------------
// GCN forward on gfx1250 (MI455X).
//   h   = relu(x @ enc_w + enc_b)
//   4x: hw = h @ conv_w;  agg = scatter_add(hw[src]*dinv[src]*dinv[dst] -> dst)
//        (agg pre-initialized with self-loop term hw[i]*dinv[i]^2 in GEMM epilogue)
//       h  = relu(agg + conv_b + h)
//   out = h @ dec_w + dec_b      (N padded 40 -> 48, masked store)
//
// GEMM: V_WMMA_F32_16X16X4_F32 (exact fp32), persistent B fragments in VGPRs,
//       A tiles staged via GLOBAL_LOAD_ASYNC_TO_LDS_B128 (ASYNCcnt path).
// Scatter: wave-per-edge, hardware global_atomic_add_f32 into L2-resident agg.
// ---------------------------------------------------------------------------

typedef float v2f __attribute__((ext_vector_type(2)));
typedef float v8f __attribute__((ext_vector_type(8)));

#define NH 128            // feature / hidden width (K dimension of all GEMMs)
#define LDS_STRIDE 132    // 132 % 64 == 4 -> conflict-free float2 reads

// ---------------- degree / normalization ----------------

__global__ void gcn_deg_init(unsigned* __restrict__ deg, int n) {
    int i = blockIdx.x * blockDim.x + threadIdx.x;
    if (i < n) deg[i] = 1u;   // self loop
}

__global__ void gcn_deg_count(const int* __restrict__ dst, unsigned* __restrict__ deg, int e) {
    int i = blockIdx.x * blockDim.x + threadIdx.x;
    if (i < e) atomicAdd(&deg[dst[i]], 1u);
}

__global__ void gcn_dinv(const unsigned* __restrict__ deg, float* __restrict__ dinv, int n) {
    int i = blockIdx.x * blockDim.x + threadIdx.x;
    if (i < n) {
        float d = (float)deg[i];
        dinv[i] = (d > 0.f) ? rsqrtf(d) : 0.f;
    }
}

// ---------------- decoder weight/bias padding (40 -> 48 cols) ----------------

__global__ void gcn_pad_dec(const float* __restrict__ dw, const float* __restrict__ db,
                            float* __restrict__ wpad, float* __restrict__ bpad) {
    int i = blockIdx.x * blockDim.x + threadIdx.x;
    if (i < NH * 48) {
        int k = i / 48, n = i - k * 48;
        wpad[i] = (n < 40) ? dw[k * 40 + n] : 0.f;
    }
    if (i < 48) bpad[i] = (i < 40) ? db[i] : 0.f;
}

// ---------------- WMMA GEMM: out[M x ldOut] = A[M x 128] @ W[128 x Nld] ----------------
// MODE 0: out = relu(acc + bias)                       (encoder)
// MODE 1: out = acc; out2 = acc * dinv[row]^2          (conv: hw + self-loop agg init)
// MODE 2: out = acc + bias, store only col < colLimit  (decoder, padded N)
//
// Block = 256 threads = 8 wave32s; wave w owns output columns [16w, 16w+16).
// Each wave keeps all 32 K-step B fragments resident in VGPRs and grid-strides
// over 16-row M tiles; the 16x128 A tile is staged with async-to-LDS copies.

template <int MODE>
__global__ __launch_bounds__(256) void gcn_gemm_wmma(
    const float* __restrict__ A, const float* __restrict__ W,
    const float* __restrict__ bias, const float* __restrict__ dinv,
    float* __restrict__ out, float* __restrict__ out2,
    int M, int Ncols, int Nld, int ldOut, int colLimit, int mtiles)
{
    __shared__ float ldsA[16 * LDS_STRIDE];

    const int tid  = threadIdx.x;
    const int wave = tid >> 5;
    const int lane = tid & 31;
    const int half = lane >> 4;    // 0: lanes 0-15, 1: lanes 16-31
    const int idx  = lane & 15;
    const int n0   = wave * 16;
    const bool active = (n0 < Ncols);   // wave-uniform
    const int col  = n0 + idx;

    // Persistent B fragments: B[kk] covers K = {4kk+2h, 4kk+2h+1}, column `col`.
    v2f B[32];
    if (active) {
        #pragma unroll
        for (int kk = 0; kk < 32; ++kk) {
            const int k = kk * 4 + 2 * half;
            B[kk].x = W[(size_t)k       * Nld + col];
            B[kk].y = W[(size_t)(k + 1) * Nld + col];
        }
    }

    for (int t = blockIdx.x; t < mtiles; t += gridDim.x) {
        const int m0 = t * 16;

        // Stage 16x128 A tile: 512 float4 chunks, async copy into LDS.
        for (int i = tid; i < 16 * 32; i += 256) {
            const int r  = i >> 5;          // tile row 0..15
            const int c4 = i & 31;          // float4 column 0..31
            int row = m0 + r;
            row = row < M ? row : M - 1;    // clamp (M % 16 == 0 in practice)
            const void* g = (const char*)A + ((size_t)row * NH + c4 * 4) * sizeof(float);
            const unsigned l = (unsigned)(uintptr_t)&ldsA[r * LDS_STRIDE + c4 * 4];
            asm volatile("global_load_async_to_lds_b128 %0, %1, off"
                         :: "v"(l), "v"(g) : "memory");
        }
        asm volatile("s_wait_asynccnt 0x0" ::: "memory");
        __syncthreads();

        if (active) {
            v8f c = {};
            #pragma unroll
            for (int kk = 0; kk < 32; ++kk) {
                v2f a = *(const v2f*)(&ldsA[idx * LDS_STRIDE + kk * 4 + 2 * half]);
                c = __builtin_amdgcn_wmma_f32_16x16x4_f32(false, a, false, B[kk],
                                                          (short)0, c, false, false);
            }

            // Epilogue: C/D layout -> VGPR r holds row m0 + r + 8*half, col = n0+idx.
            const int rbase = m0 + 8 * half;
            if (MODE == 0) {
                const float bcol = bias[col];
                float* po = out + (size_t)rbase * ldOut + col;
                #pragma unroll
                for (int r = 0; r < 8; ++r) {
                    float v = c[r] + bcol;
                    po[(size_t)r * ldOut] = v > 0.f ? v : 0.f;
                }
            } else if (MODE == 1) {
                float* po = out  + (size_t)rbase * ldOut + col;
                float* pa = out2 + (size_t)rbase * ldOut + col;
                #pragma unroll
                for (int r = 0; r < 8; ++r) {
                    const float v  = c[r];
                    const float di = dinv[rbase + r];
                    po[(size_t)r * ldOut] = v;
                    pa[(size_t)r * ldOut] = v * di * di;   // self-loop term
                }
            } else { // MODE 2
                if (col < colLimit) {
                    const float bcol = bias[col];
                    float* po = out + (size_t)rbase * ldOut + col;
                    #pragma unroll
                    for (int r = 0; r < 8; ++r)
                        po[(size_t)r * ldOut] = c[r] + bcol;
                }
            }
        }
        __syncthreads();   // LDS tile free before next stage
    }
}

// ---------------- edge scatter: agg[dst] += hw[src] * dinv[src]*dinv[dst] ----------------
// One wave per edge; lane handles features {lane, lane+32, lane+64, lane+96}.

__global__ __launch_bounds__(256) void gcn_scatter(
    const int* __restrict__ src, const int* __restrict__ dst,
    const float* __restrict__ dinv, const float* __restrict__ hw,
    float* __restrict__ agg, int E)
{
    const int lane = threadIdx.x & 31;
    const int w    = blockIdx.x * (blockDim.x >> 5) + (threadIdx.x >> 5);
    const int nw   = gridDim.x * (blockDim.x >> 5);

    for (int e = w; e < E; e += nw) {
        const int s = src[e], d = dst[e];
        const float nrm = dinv[s] * dinv[d];
        const float* hr = hw  + (size_t)s * NH;
        float*       ar = agg + (size_t)d * NH;
        #pragma unroll
        for (int j = 0; j < 4; ++j) {
            const int f = lane + 32 * j;
            unsafeAtomicAdd(&ar[f], hr[f] * nrm);   // hardware global_atomic_add_f32
        }
    }
}

// ---------------- h = relu(agg + conv_b + h), float4-vectorized ----------------

__global__ void gcn_residual_relu(float4* __restrict__ h, const float4* __restrict__ agg,
                                  const float4* __restrict__ bias, int total4)
{
    int i = blockIdx.x * blockDim.x + threadIdx.x;
    if (i < total4) {
        float4 a = agg[i], b = bias[i & 31], hh = h[i];
        float4 r;
        r.x = a.x + b.x + hh.x; r.x = r.x > 0.f ? r.x : 0.f;
        r.y = a.y + b.y + hh.y; r.y = r.y > 0.f ? r.y : 0.f;
        r.z = a.z + b.z + hh.z; r.z = r.z > 0.f ? r.z : 0.f;
        r.w = a.w + b.w + hh.w; r.w = r.w > 0.f ? r.w : 0.f;
        h[i] = r;
    }
}

// ---------------------------------------------------------------------------

extern "C" void kernel_launch(void* const* d_in, const int* in_sizes, int n_in,
                              void* d_out, int out_size, void* d_ws, size_t ws_size,
                              hipStream_t stream) {
    const float* x      = (const float*)d_in[0];
    const int*   ei     = (const int*)d_in[1];
    const float* enc_w  = (const float*)d_in[2];
    const float* enc_b  = (const float*)d_in[3];
    const float* conv_w = (const float*)d_in[4];
    const float* conv_b = (const float*)d_in[5];
    const float* dec_w  = (const float*)d_in[6];
    const float* dec_b  = (const float*)d_in[7];

    const int N = in_sizes[0] / NH;      // 100000
    const int E = in_sizes[1] / 2;       // 1600000
    const int* src = ei;
    const int* dst = ei + E;

    // workspace layout
    char* ws = (char*)d_ws;
    const size_t HB = (size_t)N * NH * sizeof(float);   // 51.2 MB each
    const size_t NB = (size_t)N * sizeof(float);
    float*    h    = (float*)(ws);
    float*    hw   = (float*)(ws + HB);
    float*    agg  = (float*)(ws + 2 * HB);
    float*    dinv = (float*)(ws + 3 * HB);
    unsigned* deg  = (unsigned*)(ws + 3 * HB + NB);
    float*    wpad = (float*)(ws + 3 * HB + 2 * NB);
    float*    bpad = wpad + NH * 48;

    const int mtiles = (N + 15) / 16;
    const int ggrid  = mtiles < 1024 ? mtiles : 1024;   // persistent-B grid-stride

    // normalization: deg -> dinv
    gcn_deg_init <<<(N + 255) / 256, 256, 0, stream>>>(deg, N);
    gcn_deg_count<<<(E + 255) / 256, 256, 0, stream>>>(dst, deg, E);
    gcn_dinv     <<<(N + 255) / 256, 256, 0, stream>>>(deg, dinv, N);
    gcn_pad_dec  <<<(NH * 48 + 255) / 256, 256, 0, stream>>>(dec_w, dec_b, wpad, bpad);

    // encoder: h = relu(x @ enc_w + enc_b)
    gcn_gemm_wmma<0><<<ggrid, 256, 0, stream>>>(x, enc_w, enc_b, nullptr,
                                                h, nullptr, N, NH, NH, NH, NH, mtiles);

    // 4 GCN layers
    for (int l = 0; l < 4; ++l) {
        gcn_gemm_wmma<1><<<ggrid, 256, 0, stream>>>(h, conv_w, nullptr, dinv,
                                                    hw, agg, N, NH, NH, NH, NH, mtiles);
        gcn_scatter<<<4096, 256, 0, stream>>>(src, dst, dinv, hw, agg, E);
        gcn_residual_relu<<<(N * 32 + 255) / 256, 256, 0, stream>>>(
            (float4*)h, (const float4*)agg, (const float4*)conv_b, N * 32);
    }

    // decoder: out = h @ dec_w + dec_b  (padded to 48 cols, store col < 40)
    gcn_gemm_wmma<2><<<ggrid, 256, 0, stream>>>(h, wpad, bpad, nullptr,
                                                (float*)d_out, nullptr,
                                                N, 48, 48, 40, 40, mtiles);
}